// ViTCoMer_71536975282225
// MI455X (gfx1250) — compile-verified
//
#include <hip/hip_runtime.h>
#include <hip/hip_bf16.h>

typedef __attribute__((ext_vector_type(2))) float v2f;
typedef __attribute__((ext_vector_type(8))) float v8f;

#define GEMM_TM 128
#define GEMM_TN 64
#define GEMM_TK 32

// ---------------------------------------------------------------------------
// CDNA5 async global->LDS helpers (ASYNCcnt-tracked, bypasses VGPR staging)
// ---------------------------------------------------------------------------
__device__ __forceinline__ void async_load_b128(unsigned lds_off, const void* gaddr)
{
    // vdst = per-lane LDS byte address, vaddr = per-lane 64-bit global address
    asm volatile("global_load_async_to_lds_b128 %0, %1, off"
                 :: "v"(lds_off), "v"(gaddr)
                 : "memory");
}
__device__ __forceinline__ void wait_asynccnt0()
{
    asm volatile("s_wait_asynccnt 0x0" ::: "memory");
}
__device__ __forceinline__ unsigned lds_addr(const void* p)
{
    // generic LDS pointer: low 32 bits are the LDS byte offset
    return (unsigned)(uintptr_t)p;
}

// ---------------------------------------------------------------------------
// Async double-buffered f32 GEMM + bias: Y[M,N] = X[M,K] @ W[K,N] + bias[N]
// Requirements: M % 128 == 0, N % 64 == 0, K % 32 == 0.
// global_load_async_to_lds_b128 overlaps next K-chunk with V_WMMA_F32_16X16X4_F32.
// ---------------------------------------------------------------------------
__global__ __launch_bounds__(256) void gemm_bias_f32_wmma_async(
    const float* __restrict__ X, const float* __restrict__ W,
    const float* __restrict__ bias, float* __restrict__ Y,
    int M, int N, int K)
{
    __shared__ __align__(16) float As[2][GEMM_TM][GEMM_TK + 4]; // stride 36
    __shared__ __align__(16) float Bs[2][GEMM_TK][GEMM_TN + 4]; // stride 68

    const int tid  = threadIdx.x;
    const int lane = tid & 31;
    const int wave = tid >> 5;
    const int lm   = lane & 15;
    const int lk2  = lane >> 4;

    const int m0 = blockIdx.y * GEMM_TM;
    const int n0 = blockIdx.x * GEMM_TN;

    const int wm = (wave & 3) * 32;
    const int wn = (wave >> 2) * 32;

    v8f acc[2][2] = {};

    auto issue_tile = [&](int buf, int k0) {
        // A tile: 128x32 floats = 1024 x b128, 4 per thread
        #pragma unroll
        for (int i = 0; i < 4; ++i) {
            int item = tid + 256 * i;
            int row  = item >> 3;              // 8 x float4 per row
            int c4   = (item & 7) * 4;
            async_load_b128(lds_addr(&As[buf][row][c4]),
                            X + (size_t)(m0 + row) * K + (k0 + c4));
        }
        // B tile: 32x64 floats = 512 x b128, 2 per thread
        #pragma unroll
        for (int i = 0; i < 2; ++i) {
            int item = tid + 256 * i;
            int row  = item >> 4;              // 16 x float4 per row
            int c4   = (item & 15) * 4;
            async_load_b128(lds_addr(&Bs[buf][row][c4]),
                            W + (size_t)(k0 + row) * N + (n0 + c4));
        }
    };

    const int nchunks = K / GEMM_TK;
    issue_tile(0, 0);

    for (int c = 0; c < nchunks; ++c) {
        const int buf = c & 1;
        wait_asynccnt0();     // our async writes for buf are in LDS
        __syncthreads();      // all waves' writes visible; prior readers done
        if (c + 1 < nchunks)
            issue_tile(buf ^ 1, (c + 1) * GEMM_TK);   // overlap with compute

        #pragma unroll
        for (int kk = 0; kk < GEMM_TK; kk += 4) {
            const int kb = kk + 2 * lk2;
            v2f a[2], b[2];
            #pragma unroll
            for (int t = 0; t < 2; ++t) {
                a[t]   = *(const v2f*)&As[buf][wm + t * 16 + lm][kb];
                b[t].x = Bs[buf][kb    ][wn + t * 16 + lm];
                b[t].y = Bs[buf][kb + 1][wn + t * 16 + lm];
            }
            #pragma unroll
            for (int i = 0; i < 2; ++i)
                #pragma unroll
                for (int j = 0; j < 2; ++j)
                    acc[i][j] = __builtin_amdgcn_wmma_f32_16x16x4_f32(
                        false, a[i], false, b[j], (short)0, acc[i][j], false, false);
        }
        // no trailing barrier: next iteration's barrier orders buffer reuse
    }

    #pragma unroll
    for (int i = 0; i < 2; ++i) {
        #pragma unroll
        for (int j = 0; j < 2; ++j) {
            int n = n0 + wn + j * 16 + lm;
            float bb = bias[n];
            #pragma unroll
            for (int r = 0; r < 8; ++r) {
                int m = m0 + wm + i * 16 + r + 8 * lk2;
                Y[(size_t)m * N + n] = acc[i][j][r] + bb;
            }
        }
    }
}

// ---------------------------------------------------------------------------
// Synchronous guarded f32 GEMM + bias (N arbitrary) for the narrow projections.
// ---------------------------------------------------------------------------
__global__ __launch_bounds__(256) void gemm_bias_f32_wmma(
    const float* __restrict__ X, const float* __restrict__ W,
    const float* __restrict__ bias, float* __restrict__ Y,
    int M, int N, int K)
{
    __shared__ __align__(16) float As[GEMM_TM][GEMM_TK + 4];
    __shared__ __align__(16) float Bs[GEMM_TK][GEMM_TN + 4];

    const int tid  = threadIdx.x;
    const int lane = tid & 31;
    const int wave = tid >> 5;
    const int lm   = lane & 15;
    const int lk2  = lane >> 4;

    const int m0 = blockIdx.y * GEMM_TM;
    const int n0 = blockIdx.x * GEMM_TN;

    const int wm = (wave & 3) * 32;
    const int wn = (wave >> 2) * 32;

    v8f acc[2][2] = {};

    for (int k0 = 0; k0 < K; k0 += GEMM_TK) {
        #pragma unroll
        for (int i = 0; i < 4; ++i) {
            int item = tid + 256 * i;
            int row  = item >> 3;
            int c4   = (item & 7) * 4;
            float4 val = *(const float4*)(X + (size_t)(m0 + row) * K + (k0 + c4));
            *(float4*)&As[row][c4] = val;
        }
        #pragma unroll
        for (int i = 0; i < 8; ++i) {
            int item = tid + 256 * i;
            int row  = item >> 6;
            int col  = item & 63;
            int n    = n0 + col;
            Bs[row][col] = (n < N) ? W[(size_t)(k0 + row) * N + n] : 0.0f;
        }
        __syncthreads();

        #pragma unroll
        for (int kk = 0; kk < GEMM_TK; kk += 4) {
            const int kb = kk + 2 * lk2;
            v2f a[2], b[2];
            #pragma unroll
            for (int t = 0; t < 2; ++t) {
                a[t]   = *(const v2f*)&As[wm + t * 16 + lm][kb];
                b[t].x = Bs[kb    ][wn + t * 16 + lm];
                b[t].y = Bs[kb + 1][wn + t * 16 + lm];
            }
            #pragma unroll
            for (int i = 0; i < 2; ++i)
                #pragma unroll
                for (int j = 0; j < 2; ++j)
                    acc[i][j] = __builtin_amdgcn_wmma_f32_16x16x4_f32(
                        false, a[i], false, b[j], (short)0, acc[i][j], false, false);
        }
        __syncthreads();
    }

    #pragma unroll
    for (int i = 0; i < 2; ++i) {
        #pragma unroll
        for (int j = 0; j < 2; ++j) {
            int n = n0 + wn + j * 16 + lm;
            if (n < N) {
                float bb = bias[n];
                #pragma unroll
                for (int r = 0; r < 8; ++r) {
                    int m = m0 + wm + i * 16 + r + 8 * lk2;
                    Y[(size_t)m * N + n] = acc[i][j][r] + bb;
                }
            }
        }
    }
}

// ---------------------------------------------------------------------------
// Softmax over contiguous groups of 12 (in-place). rows = B*LQ*NH.
// ---------------------------------------------------------------------------
__global__ __launch_bounds__(256) void softmax12_kernel(float* __restrict__ a, int rows)
{
    int r = blockIdx.x * blockDim.x + threadIdx.x;
    if (r >= rows) return;
    float* p = a + (size_t)r * 12;
    float mx = p[0];
    #pragma unroll
    for (int i = 1; i < 12; ++i) mx = fmaxf(mx, p[i]);
    float s = 0.f;
    float e[12];
    #pragma unroll
    for (int i = 0; i < 12; ++i) { e[i] = __expf(p[i] - mx); s += e[i]; }
    float inv = 1.0f / s;
    #pragma unroll
    for (int i = 0; i < 12; ++i) p[i] = e[i] * inv;
}

// ---------------------------------------------------------------------------
// Reference points: ref[b,l,:] = (mean(rp[:2304]) + rp[2304+l] + bilinear12to24)/3
// ---------------------------------------------------------------------------
__global__ __launch_bounds__(576) void refpoints_kernel(
    const float* __restrict__ rp, float* __restrict__ ref)
{
    const int b = blockIdx.x;
    const int t = threadIdx.x;
    __shared__ float sx[576];
    __shared__ float sy[576];
    __shared__ float p1[2];

    const float* rpb = rp + (size_t)b * 3024 * 2;

    float ax = 0.f, ay = 0.f;
    for (int i = t; i < 2304; i += 576) { ax += rpb[i * 2]; ay += rpb[i * 2 + 1]; }
    sx[t] = ax; sy[t] = ay;
    __syncthreads();
    if (t < 64) {
        float axx = 0.f, ayy = 0.f;
        for (int j = t; j < 576; j += 64) { axx += sx[j]; ayy += sy[j]; }
        sx[t] = axx; sy[t] = ayy;
    }
    __syncthreads();
    if (t == 0) {
        float axx = 0.f, ayy = 0.f;
        for (int j = 0; j < 64; ++j) { axx += sx[j]; ayy += sy[j]; }
        p1[0] = axx * (1.0f / 2304.0f);
        p1[1] = ayy * (1.0f / 2304.0f);
    }
    __syncthreads();

    const int oy = t / 24, ox = t % 24;
    float ysf = fminf(fmaxf(0.5f * (float)oy - 0.25f, 0.f), 11.f);
    float xsf = fminf(fmaxf(0.5f * (float)ox - 0.25f, 0.f), 11.f);
    int y0 = (int)floorf(ysf); int y1 = min(y0 + 1, 11);
    int x0 = (int)floorf(xsf); int x1 = min(x0 + 1, 11);
    float wy = ysf - (float)y0, wx = xsf - (float)x0;
    const float* p3 = rpb + 2880 * 2;
    #pragma unroll
    for (int c = 0; c < 2; ++c) {
        float v00 = p3[(y0 * 12 + x0) * 2 + c];
        float v01 = p3[(y0 * 12 + x1) * 2 + c];
        float v10 = p3[(y1 * 12 + x0) * 2 + c];
        float v11 = p3[(y1 * 12 + x1) * 2 + c];
        float pv  = v00 * (1 - wy) * (1 - wx) + v01 * (1 - wy) * wx
                  + v10 * wy * (1 - wx)       + v11 * wy * wx;
        float p2  = rpb[(2304 + t) * 2 + c];
        ref[((size_t)b * 576 + t) * 2 + c] = (p1[c] + p2 + pv) * (1.0f / 3.0f);
    }
}

// ---------------------------------------------------------------------------
// Deformable sampling: one wave per (b,l,h); lane carries channels {2*lane, 2*lane+1}.
// ---------------------------------------------------------------------------
__global__ __launch_bounds__(256) void msda_sample_kernel(
    const float* __restrict__ v, const float* __restrict__ off,
    const float* __restrict__ aw, const float* __restrict__ ref,
    float* __restrict__ attn)
{
    const int gwave = (int)((blockIdx.x * blockDim.x + threadIdx.x) >> 5);
    const int lane  = threadIdx.x & 31;
    const int h = gwave % 6;
    const int l = (gwave / 6) % 576;
    const int b = gwave / (6 * 576);
    const size_t bl = (size_t)b * 576 + l;

    const float rx = ref[bl * 2 + 0];
    const float ry = ref[bl * 2 + 1];
    const float* offp = off + bl * 144 + h * 24;
    const float* awp  = aw  + bl * 72  + h * 12;

    const int Hs[3] = {48, 24, 12};
    const int Ss[3] = {0, 2304, 2880};

    float a0 = 0.f, a1 = 0.f;
    #pragma unroll
    for (int lvl = 0; lvl < 3; ++lvl) {
        const int Hd = Hs[lvl];
        #pragma unroll
        for (int p = 0; p < 4; ++p) {
            float cx = offp[(lvl * 4 + p) * 2 + 0] + rx;
            float cy = offp[(lvl * 4 + p) * 2 + 1] + ry;
            int xi = (int)floorf(cx); xi = min(max(xi, 0), Hd - 1);
            int yi = (int)floorf(cy); yi = min(max(yi, 0), Hd - 1);
            int flat = Ss[lvl] + xi * Hd + yi;
            const float* g = v + ((size_t)b * 3024 + flat) * 384 + h * 64 + lane * 2;
            float w = awp[lvl * 4 + p];
            a0 += w * g[0];
            a1 += w * g[1];
        }
    }
    float* op = attn + bl * 384 + h * 64 + lane * 2;
    op[0] = a0;
    op[1] = a1;
}

// ---------------------------------------------------------------------------
// Launch
// ---------------------------------------------------------------------------
extern "C" void kernel_launch(void* const* d_in, const int* in_sizes, int n_in,
                              void* d_out, int out_size, void* d_ws, size_t ws_size,
                              hipStream_t stream)
{
    const float* query = (const float*)d_in[0];
    const float* value = (const float*)d_in[1];
    const float* rp    = (const float*)d_in[2];
    const float* Wv    = (const float*)d_in[3];
    const float* bv    = (const float*)d_in[4];
    const float* Woff  = (const float*)d_in[5];
    const float* boff  = (const float*)d_in[6];
    const float* Wa    = (const float*)d_in[7];
    const float* ba    = (const float*)d_in[8];
    const float* Wo    = (const float*)d_in[9];
    const float* bo    = (const float*)d_in[10];
    float* out = (float*)d_out;

    float* ws   = (float*)d_ws;
    float* vprj = ws;                                     // 193536*384
    float* offb = vprj + (size_t)193536 * 384;            // 36864*144
    float* awl  = offb + (size_t)36864 * 144;             // 36864*72
    float* refb = awl  + (size_t)36864 * 72;              // 64*576*2
    float* attn = refb + (size_t)64 * 576 * 2;            // 36864*384

    // 1) value projection (async double-buffered WMMA GEMM)
    gemm_bias_f32_wmma_async<<<dim3(384 / GEMM_TN, 193536 / GEMM_TM), 256, 0, stream>>>(
        value, Wv, bv, vprj, 193536, 384, 384);

    // 2) offsets (N=144, guarded sync GEMM)
    gemm_bias_f32_wmma<<<dim3(3, 36864 / GEMM_TM), 256, 0, stream>>>(
        query, Woff, boff, offb, 36864, 144, 384);

    // 3) attention logits (N=72, guarded sync GEMM)
    gemm_bias_f32_wmma<<<dim3(2, 36864 / GEMM_TM), 256, 0, stream>>>(
        query, Wa, ba, awl, 36864, 72, 384);

    // 4) softmax
    softmax12_kernel<<<(221184 + 255) / 256, 256, 0, stream>>>(awl, 221184);

    // 5) reference points
    refpoints_kernel<<<64, 576, 0, stream>>>(rp, refb);

    // 6) deformable sampling
    msda_sample_kernel<<<221184 / 8, 256, 0, stream>>>(vprj, offb, awl, refb, attn);

    // 7) output projection (async double-buffered WMMA GEMM)
    gemm_bias_f32_wmma_async<<<dim3(384 / GEMM_TN, 36864 / GEMM_TM), 256, 0, stream>>>(
        attn, Wo, bo, out, 36864, 384, 384);
}